// _Concat_61710090109042
// MI455X (gfx1250) — compile-verified
//
#include <hip/hip_runtime.h>

// Concat(x:[32,256,64,64] f32, broadcast(y:[32,4]) -> [32,4,64,64]) on channel dim.
// Pure streaming copy: ~258 MiB of HBM traffic, ~11.6 us floor at 23.3 TB/s.
// Strategy: 128-bit vector ops (b128), non-temporal cache hints (touch-once data),
// gfx1250 global_prefetch ahead of the read stream, shift/mask index math.

// Native clang vector type (16 B) — works with __builtin_nontemporal_* and
// lowers to global_load_b128 / global_store_b128.
typedef float v4f __attribute__((ext_vector_type(4)));

// ---- compile-time geometry (reference shapes are fixed) ----
static constexpr unsigned B        = 32;
static constexpr unsigned C_X      = 256;
static constexpr unsigned C_Y      = 4;
static constexpr unsigned HW       = 64 * 64;                  // 4096
static constexpr unsigned X4_PER_B = C_X * HW / 4;             // 262144 = 2^18
static constexpr unsigned Y4_PER_B = C_Y * HW / 4;             // 4096   = 2^12
static constexpr unsigned O4_PER_B = (C_X + C_Y) * HW / 4;     // 266240
static constexpr unsigned N4_X     = B * X4_PER_B;             // 8388608 float4
static constexpr unsigned N4_Y     = B * Y4_PER_B;             // 131072  float4
static constexpr unsigned PF_DIST  = 1u << 16;                 // prefetch ~1 MiB ahead

__global__ __launch_bounds__(256) void concat_copy_x(const v4f* __restrict__ x,
                                                     v4f* __restrict__ out) {
    const unsigned i = blockIdx.x * blockDim.x + threadIdx.x;   // < N4_X
    const unsigned b   = i >> 18;                // i / X4_PER_B
    const unsigned rem = i & (X4_PER_B - 1u);

    // Speculative prefetch of the read stream ~1 MiB ahead (global_prefetch_b8).
    if (i + PF_DIST < N4_X) {
        __builtin_prefetch(&x[i + PF_DIST], 0, 0);
    }

    // Touch-once data: non-temporal load + non-temporal store (th:*_NT).
    const v4f v = __builtin_nontemporal_load(&x[i]);
    __builtin_nontemporal_store(v, &out[b * O4_PER_B + rem]);
}

__global__ __launch_bounds__(256) void concat_fill_y(const float* __restrict__ y,
                                                     v4f* __restrict__ out) {
    const unsigned i = blockIdx.x * blockDim.x + threadIdx.x;   // < N4_Y
    const unsigned b = i >> 12;                  // i / Y4_PER_B
    const unsigned j = i & (Y4_PER_B - 1u);
    const unsigned c = j >> 10;                  // j / (HW/4)  (channel 0..3)

    // Same scalar for 1024 consecutive float4s -> wave-uniform load, cache hit.
    const float s = y[(b << 2) + c];
    const v4f v = {s, s, s, s};
    __builtin_nontemporal_store(v, &out[b * O4_PER_B + X4_PER_B + j]);
}

extern "C" void kernel_launch(void* const* d_in, const int* in_sizes, int n_in,
                              void* d_out, int out_size, void* d_ws, size_t ws_size,
                              hipStream_t stream) {
    (void)in_sizes; (void)n_in; (void)d_ws; (void)ws_size; (void)out_size;

    const v4f*   x = (const v4f*)  d_in[0];   // [32,256,64,64] f32, 16B-aligned
    const float* y = (const float*)d_in[1];   // [32,4] f32
    v4f* out = (v4f*)d_out;                   // [32,260,64,64] f32

    // 8,388,608 float4 / 256 threads = 32768 blocks (8 wave32 per block).
    concat_copy_x<<<N4_X / 256, 256, 0, stream>>>(x, out);
    // 131,072 float4 / 256 threads = 512 blocks.
    concat_fill_y<<<N4_Y / 256, 256, 0, stream>>>(y, out);
}